// jslab_25812753449702
// MI455X (gfx1250) — compile-verified
//
#include <hip/hip_runtime.h>

// MI455X / gfx1250, wave32. f32 WMMA (16x16x4) used for conv1 implicit GEMM.

typedef __attribute__((ext_vector_type(2))) float v2f;
typedef __attribute__((ext_vector_type(8))) float v8f;

#define NSUB 60
#define DT_  60.0f
#define NX   256
#define NPIX (256 * 256)
#define NOUT 24          // output time slots
#define NINT 23          // intervals actually needed (last interval's result is dropped)
#define TILE 16
#define HW   18          // hidden tile width (TILE+2), conv2 needs 1-halo of hidden
#define HP   (HW * HW)   // 324 hidden pixels
#define HPAD 336         // 21 groups * 16 px (rows 324..335 are write-only scratch)
#define HSTR 17          // padded channel stride for hidden buffer (bank-friendly)
#define XW   20          // input halo tile width (TILE+4)
#define XN   (XW * XW)   // 400

// Compile-time xs/Bw1 offsets for the im2col A/B fragments.
// k in [0,18) maps to (channel c, tap dy,dx); k>=18 is K-padding where Bw1==0,
// so A may load any in-range value (clamp) -- product is zero regardless.
__device__ constexpr int xoff(int k) {
    const int kk = k > 17 ? 17 : k;
    const int c  = kk / 9;
    const int t  = kk - 9 * c;
    const int dy = t / 3;
    const int dx = t - 3 * dy;
    return c * XN + dy * XW + dx;
}
__device__ constexpr int boff(int k) { return k * 16; }

// ---------------------------------------------------------------------------
// Kernel A: for each (interval iout, 16x16 tile), compute
//   G[iout] = sum_{iin=0..59} a^{59-iin} * dt * F_iin      (complex, a = 1 - i*fc*dt)
// where F = K*TA_interp - relu(conv2(relu(conv1([Ug,Vg]_interp)))).
// Fully parallel over 23*256 blocks; G written straight into d_out.
// ---------------------------------------------------------------------------
__global__ __launch_bounds__(256) void forcing_kernel(
    const float* __restrict__ K0p, const float* __restrict__ fcp,
    const float* __restrict__ TAx, const float* __restrict__ TAy,
    const float* __restrict__ Ug,  const float* __restrict__ Vg,
    const float* __restrict__ W1,  const float* __restrict__ b1,
    const float* __restrict__ W2,  const float* __restrict__ b2,
    float* __restrict__ G)   // == d_out, layout [2][24][256][256]
{
    __shared__ float gU0[XN], gU1[XN], gV0[XN], gV1[XN]; // raw forcing frames (halo)
    __shared__ float xs[2 * XN];                         // interpolated [Ug,Vg] halo
    __shared__ float hs[HPAD * HSTR];                    // hidden, stride 17
    __shared__ float Bw1[20 * 16];                       // W1 as B-matrix [k(20)][o(16)]
    __shared__ float w2s[2 * 9 * 16];                    // W2 as [o][tap][c]

    const int tid  = threadIdx.x;
    const int lane = tid & 31;      // wave32
    const int l15  = lane & 15;
    const int wave = tid >> 5;      // 8 waves / block
    const int iout = blockIdx.y;
    const int by   = (blockIdx.x >> 4) << 4;
    const int bx   = (blockIdx.x & 15) << 4;

    // ---- stage weights into LDS ----
    for (int i = tid; i < 20 * 16; i += 256) {
        const int k = i >> 4, o = i & 15;                // Bw1[k][o] = W1[o][k], K padded 18->20
        Bw1[i] = (k < 18) ? W1[o * 18 + k] : 0.0f;
    }
    for (int i = tid; i < 288; i += 256) {
        const int o = i / 144, r = i % 144, tap = r >> 4, c = r & 15;
        w2s[i] = W2[o * 144 + c * 9 + tap];              // [o][tap][c] for contiguous c
    }
    // ---- stage both forcing frames (with 2-halo, zero outside image) ----
    for (int i = tid; i < XN; i += 256) {
        const int iy = i / XW, ix = i % XW;
        const int gy = by + iy - 2, gx = bx + ix - 2;
        const bool in = ((unsigned)gy < 256u) && ((unsigned)gx < 256u);
        float u0 = 0.f, u1 = 0.f, v0 = 0.f, v1 = 0.f;
        if (in) {
            const int p0 = iout * NPIX + gy * NX + gx;
            u0 = Ug[p0]; u1 = Ug[p0 + NPIX];
            v0 = Vg[p0]; v1 = Vg[p0 + NPIX];
        }
        gU0[i] = u0; gU1[i] = u1; gV0[i] = v0; gV1[i] = v1;
    }

    // ---- per-thread pixel state ----
    const int ty0 = tid >> 4, tx0 = tid & 15;
    const int gy  = by + ty0, gx = bx + tx0;
    const int pc  = iout * NPIX + gy * NX + gx;
    const float tax0 = TAx[pc], tax1 = TAx[pc + NPIX];
    const float tay0 = TAy[pc], tay1 = TAy[pc + NPIX];
    const float Kc  = expf(K0p[0]);
    const float fc  = fcp[0];
    const float ai  = -fc * DT_;            // a = 1 + i*ai
    float wr = DT_, wi = 0.0f;              // quadrature weight dt * a^(59-iin)
    float Gx = 0.f, Gy = 0.f;
    const float b1v = b1[l15];
    const float b20 = b2[0], b21 = b2[1];
    __syncthreads();

    for (int iin = NSUB - 1; iin >= 0; --iin) {
        __syncthreads();                    // prior iter's hs reads done
        const float aa = (float)iin / 60.0f;
        const float om = 1.0f - aa;

        // interpolate forcing into xs (800 values / 256 threads)
        for (int i = tid; i < XN; i += 256) {
            xs[i]      = om * gU0[i] + aa * gU1[i];
            xs[XN + i] = om * gV0[i] + aa * gV1[i];
        }
        __syncthreads();

        // ---- conv1 via f32 WMMA: D(16px x 16ch) = A(16px x K20) * B(K20 x 16ch)
        for (int g = wave; g < 21; g += 8) {            // 21 groups cover 324 hidden px
            const int half = lane >> 4;                 // 0: K={kb,kb+1}; 1: K={kb+2,kb+3}
            int p = g * 16 + l15;
            p = p > (HP - 1) ? (HP - 1) : p;            // clamp pad pixels (A loads only)
            const int hy = p / HW, hx = p % HW;
            const int base = hy * XW + hx;              // xs tile offset of this pixel
            v8f acc = {};
#pragma unroll
            for (int kb = 0; kb < 20; kb += 4) {
                v2f a, b;
#pragma unroll
                for (int j = 0; j < 2; ++j) {
                    const int k0 = kb + j;              // half == 0
                    const int k1 = kb + j + 2;          // half == 1
                    // constexpr offsets -> one cndmask + add per load
                    a[j] = xs[base + (half ? xoff(k1) : xoff(k0))];
                    b[j] = Bw1[l15 + (half ? boff(k1) : boff(k0))];
                }
                acc = __builtin_amdgcn_wmma_f32_16x16x4_f32(
                          false, a, false, b, (short)0, acc, false, false);
            }
            // D layout: reg r, halfwave h -> pixel g*16 + r + 8*h, channel l15.
            // Unconditional stores (hs padded to 336 rows); zeroing via selects.
#pragma unroll
            for (int r = 0; r < 8; ++r) {
                const int pp  = g * 16 + r + 8 * half;
                const int hyy = pp / HW, hxx = pp - HW * hyy;
                const int gyy = by + hyy - 1, gxx = bx + hxx - 1;
                float v = acc[r] + b1v;
                v = v > 0.f ? v : 0.f;                  // relu
                const bool in = ((unsigned)gyy < 256u) && ((unsigned)gxx < 256u);
                hs[pp * HSTR + l15] = in ? v : 0.f;     // SAME zero-pad for conv2
            }
        }
        __syncthreads();

        // ---- conv2 (VALU): each thread computes its own pixel ----
        float a0 = b20, a1 = b21;
#pragma unroll
        for (int dy = 0; dy < 3; ++dy)
#pragma unroll
            for (int dx = 0; dx < 3; ++dx) {
                const int hp = (ty0 + dy) * HW + (tx0 + dx);
                const float* hrow = &hs[hp * HSTR];
                const int tap = dy * 3 + dx;
                const float* w0 = &w2s[tap * 16];
                const float* w1r = &w2s[144 + tap * 16];
#pragma unroll
                for (int c = 0; c < 16; ++c) {
                    const float h = hrow[c];
                    a0 = fmaf(w0[c], h, a0);
                    a1 = fmaf(w1r[c], h, a1);
                }
            }
        const float dis0 = a0 > 0.f ? a0 : 0.f;
        const float dis1 = a1 > 0.f ? a1 : 0.f;

        // ---- forcing and weighted accumulation ----
        const float Fx = Kc * (om * tax0 + aa * tax1) - dis0;
        const float Fy = Kc * (om * tay0 + aa * tay1) - dis1;
        Gx += wr * Fx - wi * Fy;
        Gy += wr * Fy + wi * Fx;
        const float nwr = wr - wi * ai;     // w *= (1 + i*ai)
        wi = wr * ai + wi;
        wr = nwr;
    }

    G[iout * NPIX + gy * NX + gx]               = Gx;   // ch 0
    G[NOUT * NPIX + iout * NPIX + gy * NX + gx] = Gy;   // ch 1
}

// ---------------------------------------------------------------------------
// Kernel B: per-pixel 23-step complex scan Z <- a^60 * Z + G[i]; writes outputs
// (slot 0 = zeros, slot i+1 = state after interval i). Reads G from d_out then
// overwrites d_out in place (reads complete before writes per thread).
// ---------------------------------------------------------------------------
__global__ __launch_bounds__(256) void scan_kernel(
    const float* __restrict__ fcp, float* __restrict__ out)
{
    const int pix = blockIdx.x * 256 + threadIdx.x;
    const float ai = -fcp[0] * DT_;
    float Ar = 1.f, Ai = 0.f;               // a^60
#pragma unroll
    for (int i = 0; i < NSUB; ++i) {
        const float nr = Ar - Ai * ai;
        Ai = Ar * ai + Ai;
        Ar = nr;
    }
    float Gx[NINT], Gy[NINT];
#pragma unroll
    for (int i = 0; i < NINT; ++i) {
        Gx[i] = out[i * NPIX + pix];
        Gy[i] = out[NOUT * NPIX + i * NPIX + pix];
    }
    float Zr = 0.f, Zi = 0.f;
    out[pix] = 0.f;                          // slot 0 (U)
    out[NOUT * NPIX + pix] = 0.f;            // slot 0 (V)
#pragma unroll
    for (int i = 0; i < NINT; ++i) {
        const float nZr = Ar * Zr - Ai * Zi + Gx[i];
        const float nZi = Ar * Zi + Ai * Zr + Gy[i];
        Zr = nZr; Zi = nZi;
        out[(i + 1) * NPIX + pix] = Zr;
        out[NOUT * NPIX + (i + 1) * NPIX + pix] = Zi;
    }
}

extern "C" void kernel_launch(void* const* d_in, const int* in_sizes, int n_in,
                              void* d_out, int out_size, void* d_ws, size_t ws_size,
                              hipStream_t stream) {
    const float* K0  = (const float*)d_in[0];
    const float* fc  = (const float*)d_in[1];
    const float* TAx = (const float*)d_in[2];
    const float* TAy = (const float*)d_in[3];
    const float* Ug  = (const float*)d_in[4];
    const float* Vg  = (const float*)d_in[5];
    const float* W1  = (const float*)d_in[6];
    const float* b1  = (const float*)d_in[7];
    const float* W2  = (const float*)d_in[8];
    const float* b2  = (const float*)d_in[9];
    float* out = (float*)d_out;

    dim3 gridA(256, NINT);                  // 256 tiles x 23 intervals, fully parallel
    forcing_kernel<<<gridA, 256, 0, stream>>>(K0, fc, TAx, TAy, Ug, Vg,
                                              W1, b1, W2, b2, out);
    scan_kernel<<<NPIX / 256, 256, 0, stream>>>(fc, out);
}